// TaskAlignedAssigner_14542759264893
// MI455X (gfx1250) — compile-verified
//
#include <hip/hip_runtime.h>
#include <hip/hip_bf16.h>
#include <stdint.h>

#define B_    32
#define A_    8400
#define C_    80
#define M_    64
#define TOPK_ 13
#define EPS_  1e-9f
#define BG_   80

typedef __attribute__((ext_vector_type(2))) float        v2f;
typedef __attribute__((ext_vector_type(8))) float        v8f;
typedef __attribute__((ext_vector_type(4))) unsigned int v4u;
typedef __attribute__((ext_vector_type(8))) unsigned int v8u;

// ---------------------------------------------------------------------------
// Kernel 0: zero-fill assigned_scores region (float4 coalesced)
// ---------------------------------------------------------------------------
__global__ void __launch_bounds__(256) zero_kernel(float4* __restrict__ p, int n4) {
    int i = blockIdx.x * 256 + threadIdx.x;
    if (i < n4) p[i] = make_float4(0.f, 0.f, 0.f, 0.f);
}

// ---------------------------------------------------------------------------
// Kernel 1: is_in_gts via V_WMMA_F32_16X16X4_F32.
// d = min(px-xmin, py-ymin, xmax-px, ymax-py) ; each term is affine in
// (1, px, py): D(16x16) = Coef(16x4) x Basis(4x16). One wave handles one
// 16-anchor tile across all 4 gt-tiles (M=64) and emits a 64-bit mask/anchor.
// A-frag layout: lane = half*16+row, holds A[row][2*half], A[row][2*half+1].
// B-frag layout: lane = half*16+col, holds B[2*half][col], B[2*half+1][col].
// D layout: lane VGPR r -> (m_local = r + 8*half, n = lane%16).
// ---------------------------------------------------------------------------
__global__ void __launch_bounds__(256) ingts_wmma_kernel(
    const float*  __restrict__ anchor_points,   // (A,2)
    const float4* __restrict__ gt_bboxes,       // (B,M) of float4
    uint2*        __restrict__ ingts)           // (B,A) 64-bit masks
{
    const int b    = blockIdx.y;
    const int wave = threadIdx.x >> 5;
    const int lane = threadIdx.x & 31;
    const int half = lane >> 4;
    const int col  = lane & 15;
    const int a0   = (blockIdx.x * 8 + wave) * 16;
    const int a    = a0 + col;
    const int aa   = (a < A_) ? a : (A_ - 1);

    const float px = anchor_points[aa * 2 + 0];
    const float py = anchor_points[aa * 2 + 1];

    // Basis rows: K0=1, K1=px, K2=py, K3=0
    v2f bf;
    bf.x = (half == 0) ? 1.0f : py;
    bf.y = (half == 0) ? px   : 0.0f;

    unsigned mlo = 0u, mhi = 0u;
    for (int mt = 0; mt < 4; ++mt) {
        const int m = mt * 16 + col;
        const float4 g = gt_bboxes[b * M_ + m];
        // Term coefficient rows (c0,c1,c2,c3); lane takes (c[2h], c[2h+1]).
        v2f af0, af1, af2, af3;
        af0.x = (half == 0) ? -g.x :  0.0f;  af0.y = (half == 0) ? 1.0f : 0.0f; // px - xmin
        af1.x = (half == 0) ? -g.y :  1.0f;  af1.y = 0.0f;                      // py - ymin
        af2.x = (half == 0) ?  g.z :  0.0f;  af2.y = (half == 0) ? -1.0f : 0.0f;// xmax - px
        af3.x = (half == 0) ?  g.w : -1.0f;  af3.y = 0.0f;                      // ymax - py

        v8f z = {};
        v8f d0 = __builtin_amdgcn_wmma_f32_16x16x4_f32(false, af0, false, bf, (short)0, z, false, false);
        v8f d1 = __builtin_amdgcn_wmma_f32_16x16x4_f32(false, af1, false, bf, (short)0, z, false, false);
        v8f d2 = __builtin_amdgcn_wmma_f32_16x16x4_f32(false, af2, false, bf, (short)0, z, false, false);
        v8f d3 = __builtin_amdgcn_wmma_f32_16x16x4_f32(false, af3, false, bf, (short)0, z, false, false);

        unsigned bits = 0u;
#pragma unroll
        for (int r = 0; r < 8; ++r) {
            float dm = fminf(fminf(d0[r], d1[r]), fminf(d2[r], d3[r]));
            bits |= (dm > EPS_ ? 1u : 0u) << r;
        }
        const int pos = mt * 16 + half * 8;      // bit base in the 64-bit mask
        if (pos < 32) mlo |= bits << pos;
        else          mhi |= bits << (pos - 32);
    }
    // Column n's bits live on lanes n and n+16: OR across halves.
    mlo |= (unsigned)__shfl_xor((int)mlo, 16, 32);
    mhi |= (unsigned)__shfl_xor((int)mhi, 16, 32);
    if (half == 0 && a < A_) ingts[b * A_ + a] = make_uint2(mlo, mhi);
}

// ---------------------------------------------------------------------------
// Kernel 2: per-(b,m) top-13 of metrics = score(a,label_m) * iou^6 * in_gts.
// 13 iterations of (max value, min index) masked argmax == stable lax.top_k.
// ---------------------------------------------------------------------------
__global__ void __launch_bounds__(256) topk_kernel(
    const float*  __restrict__ pred_scores,   // (B,A,C)
    const float4* __restrict__ pred_bboxes,   // (B,A)
    const float4* __restrict__ gt_bboxes,     // (B,M)
    const int*    __restrict__ gt_labels,     // (B,M)
    const uint2*  __restrict__ ingts,         // (B,A)
    int*          __restrict__ topk_out)      // (B,M,13)
{
    const int bm  = blockIdx.x;
    const int b   = bm >> 6, m = bm & (M_ - 1);
    const int tid = threadIdx.x;

    const float4 g = gt_bboxes[b * M_ + m];
    const float area1 = fmaxf(g.z - g.x, 0.f) * fmaxf(g.w - g.y, 0.f);
    const int lab = gt_labels[b * M_ + m];

    __shared__ int   sel[TOPK_];
    __shared__ float rv[256];
    __shared__ int   ri[256];

    for (int t = 0; t < TOPK_; ++t) {
        float best = -1.f; int bidx = 0x7fffffff;
        for (int a = tid; a < A_; a += 256) {
            bool used = false;
            for (int j = 0; j < t; ++j) used |= (sel[j] == a);
            if (used) continue;
            const uint2 gw = ingts[b * A_ + a];
            const unsigned bit = (((m < 32) ? gw.x : gw.y) >> (m & 31)) & 1u;
            float metric = 0.f;
            if (bit) {
                const float4 pb = pred_bboxes[b * A_ + a];
                const float x1 = fmaxf(g.x, pb.x), y1 = fmaxf(g.y, pb.y);
                const float x2 = fminf(g.z, pb.z), y2 = fminf(g.w, pb.w);
                const float ov = fmaxf(x2 - x1, 0.f) * fmaxf(y2 - y1, 0.f);
                const float area2 = fmaxf(pb.z - pb.x, 0.f) * fmaxf(pb.w - pb.y, 0.f);
                const float iou = ov / (area1 + area2 - ov + EPS_);
                const float s = pred_scores[(size_t)(b * A_ + a) * C_ + lab];
                const float i2 = iou * iou;
                metric = s * i2 * i2 * i2;
            }
            if (metric > best || (metric == best && a < bidx)) { best = metric; bidx = a; }
        }
        rv[tid] = best; ri[tid] = bidx;
        __syncthreads();
        for (int s = 128; s > 0; s >>= 1) {
            if (tid < s) {
                if (rv[tid + s] > rv[tid] || (rv[tid + s] == rv[tid] && ri[tid + s] < ri[tid])) {
                    rv[tid] = rv[tid + s]; ri[tid] = ri[tid + s];
                }
            }
            __syncthreads();
        }
        if (tid == 0) sel[t] = ri[0];
        __syncthreads();
    }
    if (tid < TOPK_) topk_out[bm * TOPK_ + tid] = sel[tid];
}

// ---------------------------------------------------------------------------
// Kernel 3: per-anchor resolve. GT boxes staged to LDS via the Tensor Data
// Mover (tensor_load_to_lds, TENSORcnt wait), descriptor per ISA 08 §8.
// Regular loads also populate the buffer (results identical) for safety.
// ---------------------------------------------------------------------------
__global__ void __launch_bounds__(256) assign_kernel(
    const float4* __restrict__ pred_bboxes,
    const float*  __restrict__ gt_bboxes_f,   // (B,M,4) flat floats
    const int*    __restrict__ gt_labels,
    const float*  __restrict__ pad_gt,        // (B,M)
    const int*    __restrict__ topk,          // (B,M,13)
    const uint2*  __restrict__ ingts,
    uint2*        __restrict__ mpmask,        // (B,A) final mask_positive
    float*        __restrict__ out_labels,    // (B,A) as float
    float4*       __restrict__ out_bboxes)    // (B,A)
{
    __shared__ float4 sGT[M_];
    __shared__ int    sTopk[M_ * TOPK_];
    __shared__ float  sPad[M_];

    const int b = blockIdx.y;
    const int tid = threadIdx.x;

    if ((tid >> 5) == 0) {
        // --- TDM async tile load: 256 f32 (gt_bboxes[b]) -> LDS sGT ---
        const unsigned long long ga =
            (unsigned long long)(uintptr_t)(gt_bboxes_f + (size_t)b * M_ * 4);
        const unsigned lds = (unsigned)(uintptr_t)(void*)&sGT[0];
        v4u g0; v8u g1;
        g0[0] = 1u;                                   // count=1 valid descriptor
        g0[1] = lds;                                  // lds_addr
        g0[2] = (unsigned)(ga & 0xFFFFFFFFull);       // global_addr[31:0]
        g0[3] = (unsigned)((ga >> 32) & 0x1FFFFFFull) // global_addr[56:32]
              | (2u << 30);                           // type=2 (image)
        g1[0] = (2u << 16);                           // data_size=4B, wg_mask=0
        g1[1] = (256u << 16);                         // tensor_dim0 lo16
        g1[2] = (1u << 16);                           // tensor_dim0 hi | tensor_dim1=1
        g1[3] = (256u << 16);                         // tile_dim0=256
        g1[4] = 1u;                                   // tile_dim1=1, tile_dim2=0
        g1[5] = 256u;                                 // tensor_dim0_stride lo32
        g1[6] = (256u << 16);                         // stride0 hi | stride1 lo16
        g1[7] = 0u;
        asm volatile("tensor_load_to_lds %0, %1" :: "s"(g0), "s"(g1) : "memory");
        __builtin_amdgcn_s_wait_tensorcnt(0);
    }
    // Redundant regular fill (same values) + pad/topk staging.
    for (int i = tid; i < M_; i += 256) {
        const float* p = gt_bboxes_f + ((size_t)b * M_ + i) * 4;
        sGT[i] = make_float4(p[0], p[1], p[2], p[3]);
        sPad[i] = pad_gt[b * M_ + i];
    }
    for (int i = tid; i < M_ * TOPK_; i += 256) sTopk[i] = topk[b * M_ * TOPK_ + i];
    __syncthreads();

    const int a = blockIdx.x * 256 + tid;
    if (a >= A_) return;

    const float4 pb = pred_bboxes[b * A_ + a];
    const float area2 = fmaxf(pb.z - pb.x, 0.f) * fmaxf(pb.w - pb.y, 0.f);
    const uint2 gm = ingts[b * A_ + a];

    unsigned lo = 0u, hi = 0u;
    float bestiou = -1.f; int argm = 0;
    for (int m = 0; m < M_; ++m) {
        const float4 g = sGT[m];
        const float x1 = fmaxf(g.x, pb.x), y1 = fmaxf(g.y, pb.y);
        const float x2 = fminf(g.z, pb.z), y2 = fminf(g.w, pb.w);
        const float ov = fmaxf(x2 - x1, 0.f) * fmaxf(y2 - y1, 0.f);
        const float area1 = fmaxf(g.z - g.x, 0.f) * fmaxf(g.w - g.y, 0.f);
        const float iou = ov / (area1 + area2 - ov + EPS_);
        if (iou > bestiou) { bestiou = iou; argm = m; }   // first-max like argmax
        const bool ing = (((m < 32) ? gm.x : gm.y) >> (m & 31)) & 1u;
        bool intk = false;
        for (int j = 0; j < TOPK_; ++j) intk |= (sTopk[m * TOPK_ + j] == a);
        if (intk && ing && sPad[m] > 0.f) {
            if (m < 32) lo |= 1u << m; else hi |= 1u << (m - 32);
        }
    }
    int cnt = __popc(lo) + __popc(hi);
    if (cnt > 1) {           // multi-assigned: keep only max-IoU gt
        lo = (argm < 32) ? (1u << argm) : 0u;
        hi = (argm >= 32) ? (1u << (argm - 32)) : 0u;
        cnt = 1;
    }
    int assigned = 0;
    if (cnt > 0) assigned = lo ? (__ffs((int)lo) - 1) : (__ffs((int)hi) + 31);
    const int labv = (cnt > 0) ? gt_labels[b * M_ + assigned] : BG_;

    out_labels[b * A_ + a] = (float)labv;
    out_bboxes[b * A_ + a] = sGT[assigned];
    mpmask[b * A_ + a] = make_uint2(lo, hi);
}

// ---------------------------------------------------------------------------
// Kernel 4: per-(b,m) max over positives of align and iou.
// ---------------------------------------------------------------------------
__global__ void __launch_bounds__(256) rowmax_kernel(
    const float*  __restrict__ pred_scores,
    const float4* __restrict__ pred_bboxes,
    const float4* __restrict__ gt_bboxes,
    const int*    __restrict__ gt_labels,
    const uint2*  __restrict__ mpmask,
    float*        __restrict__ maxmet,
    float*        __restrict__ maxiou)
{
    const int bm = blockIdx.x;
    const int b = bm >> 6, m = bm & (M_ - 1);
    const int tid = threadIdx.x;
    const float4 g = gt_bboxes[b * M_ + m];
    const float area1 = fmaxf(g.z - g.x, 0.f) * fmaxf(g.w - g.y, 0.f);
    const int lab = gt_labels[b * M_ + m];

    float lm = 0.f, li = 0.f;
    for (int a = tid; a < A_; a += 256) {
        const uint2 mm = mpmask[b * A_ + a];
        const unsigned bit = (((m < 32) ? mm.x : mm.y) >> (m & 31)) & 1u;
        if (!bit) continue;
        const float4 pb = pred_bboxes[b * A_ + a];
        const float x1 = fmaxf(g.x, pb.x), y1 = fmaxf(g.y, pb.y);
        const float x2 = fminf(g.z, pb.z), y2 = fminf(g.w, pb.w);
        const float ov = fmaxf(x2 - x1, 0.f) * fmaxf(y2 - y1, 0.f);
        const float area2 = fmaxf(pb.z - pb.x, 0.f) * fmaxf(pb.w - pb.y, 0.f);
        const float iou = ov / (area1 + area2 - ov + EPS_);
        const float s = pred_scores[(size_t)(b * A_ + a) * C_ + lab];
        const float i2 = iou * iou;
        lm = fmaxf(lm, s * i2 * i2 * i2);
        li = fmaxf(li, iou);
    }
    __shared__ float sm_[256], si_[256];
    sm_[tid] = lm; si_[tid] = li;
    __syncthreads();
    for (int s = 128; s > 0; s >>= 1) {
        if (tid < s) { sm_[tid] = fmaxf(sm_[tid], sm_[tid + s]); si_[tid] = fmaxf(si_[tid], si_[tid + s]); }
        __syncthreads();
    }
    if (tid == 0) { maxmet[bm] = sm_[0]; maxiou[bm] = si_[0]; }
}

// ---------------------------------------------------------------------------
// Kernel 5: normalized metric scatter into one-hot scores.
// ---------------------------------------------------------------------------
__global__ void __launch_bounds__(256) scores_kernel(
    const float*  __restrict__ pred_scores,
    const float4* __restrict__ pred_bboxes,
    const float4* __restrict__ gt_bboxes,
    const int*    __restrict__ gt_labels,
    const uint2*  __restrict__ mpmask,
    const float*  __restrict__ maxmet,
    const float*  __restrict__ maxiou,
    const float*  __restrict__ out_labels,
    float*        __restrict__ out_scores)
{
    const int idx = blockIdx.x * 256 + threadIdx.x;
    if (idx >= B_ * A_) return;
    const int b = idx / A_;
    const int lab = (int)out_labels[idx];
    if (lab >= BG_) return;                      // background: row stays zero

    const uint2 mm = mpmask[idx];
    const float4 pb = pred_bboxes[idx];
    const float area2 = fmaxf(pb.z - pb.x, 0.f) * fmaxf(pb.w - pb.y, 0.f);
    float factor = 0.f;
    unsigned lo = mm.x, hi = mm.y;
    while (lo | hi) {
        int m;
        if (lo) { m = __ffs((int)lo) - 1; lo &= lo - 1; }
        else    { m = __ffs((int)hi) + 31; hi &= hi - 1; }
        const float4 g = gt_bboxes[b * M_ + m];
        const float x1 = fmaxf(g.x, pb.x), y1 = fmaxf(g.y, pb.y);
        const float x2 = fminf(g.z, pb.z), y2 = fminf(g.w, pb.w);
        const float ov = fmaxf(x2 - x1, 0.f) * fmaxf(y2 - y1, 0.f);
        const float area1 = fmaxf(g.z - g.x, 0.f) * fmaxf(g.w - g.y, 0.f);
        const float iou = ov / (area1 + area2 - ov + EPS_);
        const float s = pred_scores[(size_t)idx * C_ + gt_labels[b * M_ + m]];
        const float i2 = iou * iou;
        const float al = s * i2 * i2 * i2;
        const float val = al / (maxmet[b * M_ + m] + EPS_) * maxiou[b * M_ + m];
        factor = fmaxf(factor, val);
    }
    out_scores[(size_t)idx * C_ + lab] = factor;
}

// ---------------------------------------------------------------------------
extern "C" void kernel_launch(void* const* d_in, const int* in_sizes, int n_in,
                              void* d_out, int out_size, void* d_ws, size_t ws_size,
                              hipStream_t stream) {
    (void)in_sizes; (void)n_in; (void)out_size; (void)ws_size;
    const float* pred_scores   = (const float*)d_in[0];
    const float* pred_bboxes   = (const float*)d_in[1];
    const float* anchor_points = (const float*)d_in[2];
    const int*   gt_labels     = (const int*)d_in[3];
    const float* gt_bboxes     = (const float*)d_in[4];
    const float* pad_gt        = (const float*)d_in[5];

    // scratch layout (~4.42 MB total)
    char* ws = (char*)d_ws;
    uint2* ingts  = (uint2*)ws;                                    // B*A*8
    uint2* mpmask = (uint2*)(ws + (size_t)B_ * A_ * 8);            // B*A*8
    int*   topk   = (int*)  (ws + (size_t)2 * B_ * A_ * 8);        // B*M*13*4
    float* maxmet = (float*)(ws + (size_t)2 * B_ * A_ * 8 + (size_t)B_ * M_ * TOPK_ * 4);
    float* maxiou = maxmet + B_ * M_;

    float*  out_labels = (float*)d_out;                            // B*A
    float4* out_bboxes = (float4*)((float*)d_out + (size_t)B_ * A_);
    float*  out_scores = (float*)d_out + (size_t)B_ * A_ * 5;      // B*A*C

    const int n4 = B_ * A_ * C_ / 4;
    zero_kernel<<<(n4 + 255) / 256, 256, 0, stream>>>((float4*)out_scores, n4);

    ingts_wmma_kernel<<<dim3((A_ + 127) / 128, B_), 256, 0, stream>>>(
        anchor_points, (const float4*)gt_bboxes, ingts);

    topk_kernel<<<B_ * M_, 256, 0, stream>>>(
        pred_scores, (const float4*)pred_bboxes, (const float4*)gt_bboxes,
        gt_labels, ingts, topk);

    assign_kernel<<<dim3((A_ + 255) / 256, B_), 256, 0, stream>>>(
        (const float4*)pred_bboxes, gt_bboxes, gt_labels, pad_gt, topk,
        ingts, mpmask, out_labels, out_bboxes);

    rowmax_kernel<<<B_ * M_, 256, 0, stream>>>(
        pred_scores, (const float4*)pred_bboxes, (const float4*)gt_bboxes,
        gt_labels, mpmask, maxmet, maxiou);

    scores_kernel<<<(B_ * A_ + 255) / 256, 256, 0, stream>>>(
        pred_scores, (const float4*)pred_bboxes, (const float4*)gt_bboxes,
        gt_labels, mpmask, maxmet, maxiou, out_labels, out_scores);
}